// Model_33612414059055
// MI455X (gfx1250) — compile-verified
//
#include <hip/hip_runtime.h>
#include <math.h>

typedef float v2f __attribute__((ext_vector_type(2)));
typedef float v8f __attribute__((ext_vector_type(8)));

// ------------------------------------------------------------------
// Setup: Clebsch-Gordan tables computed on device in double precision.
// Each real-CG element needs <= 8 su2_cg evaluations because each
// column of the complex->real change-of-basis q(l) has <= 2 nonzeros.
// ------------------------------------------------------------------
struct Cplx { double re, im; };
__device__ __forceinline__ Cplx cmul(Cplx a, Cplx b) {
    return { a.re * b.re - a.im * b.im, a.re * b.im + a.im * b.re };
}
__device__ __forceinline__ int imin_(int a, int b) { return a < b ? a : b; }
__device__ __forceinline__ int imax_(int a, int b) { return a > b ? a : b; }
__device__ __forceinline__ int iabs_(int a) { return a < 0 ? -a : a; }

__device__ double dfact(int n) { double r = 1.0; for (int t = 2; t <= n; ++t) r *= (double)t; return r; }

__device__ double su2_cg(int j1, int j2, int j3, int m1, int m2, int m3) {
    if (m3 != m1 + m2) return 0.0;
    int vmin = imax_(imax_(-j1 + j2 + m3, -j1 + m1), 0);
    int vmax = imin_(imin_(j2 + j3 + m1, j3 - j1 + j2), j3 + m3);
    double C = sqrt((double)(2 * j3 + 1)
        * dfact(j3 + j1 - j2) * dfact(j3 - j1 + j2) * dfact(j1 + j2 - j3) / dfact(j1 + j2 + j3 + 1)
        * dfact(j3 + m3) * dfact(j3 - m3)
        / (dfact(j1 - m1) * dfact(j1 + m1) * dfact(j2 - m2) * dfact(j2 + m2)));
    double S = 0.0;
    for (int v = vmin; v <= vmax; ++v) {
        double t = dfact(j2 + j3 + m1 - v) * dfact(j1 - m1 + v)
                 / dfact(v) / dfact(j3 - j1 + j2 - v) / dfact(j3 + m3 - v) / dfact(v + j1 - j2 - m3);
        S += ((v + j2 + m2) & 1) ? -t : t;
    }
    return C * S;
}

// q(l)[row,col] including the (-i)^l phase.
__device__ Cplx q_entry(int l, int row, int col) {
    const double RS = 0.70710678118654752440;
    int m = row - l;
    double re = 0.0, im = 0.0;
    if (m == 0) { if (col == l) re = 1.0; }
    else if (m < 0) {
        if (col == l - m)      re = RS;   // l + |m|
        else if (col == l + m) im = -RS;  // l - |m|
    } else {
        double s = (m & 1) ? -1.0 : 1.0;
        if (col == l + m)      re = s * RS;
        else if (col == l - m) im = s * RS;
    }
    Cplx v = { re, im };
    switch (l & 3) {                 // multiply by (-i)^l
        case 0:  return v;
        case 1:  return {  v.im, -v.re };
        case 2:  return { -v.re, -v.im };
        default: return { -v.im,  v.re };
    }
}

// Real CG coefficient Cr[a,b,c] for irreps (l1,l2,l3); a,b,c are block-local.
__device__ double real_cg_elem(int l1, int l2, int l3, int a, int b, int c) {
    int r1[2] = { a, 2 * l1 - a }; int n1 = (r1[1] == r1[0]) ? 1 : 2;
    int r2[2] = { b, 2 * l2 - b }; int n2 = (r2[1] == r2[0]) ? 1 : 2;
    int r3[2] = { c, 2 * l3 - c }; int n3 = (r3[1] == r3[0]) ? 1 : 2;
    double accre = 0.0;
    for (int ii = 0; ii < n1; ++ii) {
        Cplx q1 = q_entry(l1, r1[ii], a);
        if (q1.re == 0.0 && q1.im == 0.0) continue;
        for (int kk = 0; kk < n2; ++kk) {
            Cplx q2 = q_entry(l2, r2[kk], b);
            if (q2.re == 0.0 && q2.im == 0.0) continue;
            Cplx q12 = cmul(q1, q2);
            for (int nn = 0; nn < n3; ++nn) {
                Cplx q3 = q_entry(l3, r3[nn], c); q3.im = -q3.im; // conj
                if (q3.re == 0.0 && q3.im == 0.0) continue;
                double cg = su2_cg(l1, l2, l3, r1[ii] - l1, r2[kk] - l2, r3[nn] - l3);
                if (cg == 0.0) continue;
                Cplx t = cmul(q12, q3);
                accre += t.re * cg;
            }
        }
    }
    return accre;
}

__device__ __forceinline__ int lof(int idx) { int l = 0; while ((l + 1) * (l + 1) <= idx) ++l; return l; }

// Builds:
//  A1[9][16]  : A1[j][k] = sum_i K1[i,j,k]           (layer-1 collapsed matrix)
//  Bswz       : K2 flat (144x36 padded to 144x48) pre-swizzled into the
//               f32 16x16x4 WMMA B lane layout:
//               Bswz[((step*3+nt)*32 + lane)*2 + h] = K2f[4*step + 2*(lane>>4) + h][nt*16 + (lane&15)]
__global__ void setup_tables(float* __restrict__ A1, float* __restrict__ Bswz) {
    int tid = blockIdx.x * blockDim.x + threadIdx.x;
    const int NB = 36 * 3 * 32 * 2; // 6912
    if (tid < NB) {
        int h    = tid & 1;
        int lane = (tid >> 1) & 31;
        int nt   = (tid >> 6) % 3;
        int step = tid / 192;
        int m = 4 * step + ((lane >> 4) << 1) + h;   // 0..143
        int k = nt * 16 + (lane & 15);               // 0..47
        float val = 0.f;
        if (k < 36) {
            int i = m / 9, j = m - 9 * (m / 9);
            int l1 = lof(i), l2 = lof(j), l3 = lof(k);
            if (l3 >= iabs_(l1 - l2) && l3 <= l1 + l2)
                val = (float)real_cg_elem(l1, l2, l3, i - l1 * l1, j - l2 * l2, k - l3 * l3);
        }
        Bswz[((step * 3 + nt) * 32 + lane) * 2 + h] = val;
    } else if (tid < NB + 144) {
        int t = tid - NB;
        int j = t / 16, k = t % 16;
        int l2 = lof(j), l3 = lof(k);
        double s = 0.0;
        for (int l1 = 0; l1 <= 1; ++l1) {
            if (l3 < iabs_(l1 - l2) || l3 > imin_(3, l1 + l2)) continue;
            for (int i = l1 * l1; i < (l1 + 1) * (l1 + 1); ++i)
                s += real_cg_elem(l1, l2, l3, i - l1 * l1, j - l2 * l2, k - l3 * l3);
        }
        A1[j * 16 + k] = (float)s;
    }
}

// ------------------------------------------------------------------
// Per-edge real spherical harmonics (l<=2), 9 components.
// ------------------------------------------------------------------
__device__ __forceinline__ void sh_l2(float x, float y, float z, float* Y) {
    Y[0] = 0.28209479177387814f;
    Y[1] = 0.4886025119029199f * y;
    Y[2] = 0.4886025119029199f * z;
    Y[3] = 0.4886025119029199f * x;
    Y[4] = 1.0925484305920792f * x * y;
    Y[5] = 1.0925484305920792f * y * z;
    Y[6] = 0.31539156525252005f * (3.f * z * z - 1.f);
    Y[7] = 1.0925484305920792f * x * z;
    Y[8] = 0.5462742152960396f * (x * x - y * y);
}

// Layer 1 (collapsed): h1[e,k] = sum_j A1[j,k] * Y[e,j]; scatter into agg1[recv].
__global__ void __launch_bounds__(256) edge_layer1(
    const float* __restrict__ pos, const int* __restrict__ snd, const int* __restrict__ rcv,
    const float* __restrict__ A1, float* __restrict__ agg1, int E)
{
    __shared__ float sA1[144];
    if (threadIdx.x < 144) sA1[threadIdx.x] = A1[threadIdx.x];
    __syncthreads();
    int e = blockIdx.x * blockDim.x + threadIdx.x;
    if (e >= E) return;
    int s = snd[e], r = rcv[e];
    float dx = pos[3 * r + 0] - pos[3 * s + 0];
    float dy = pos[3 * r + 1] - pos[3 * s + 1];
    float dz = pos[3 * r + 2] - pos[3 * s + 2];
    float inv = 1.0f / (sqrtf(dx * dx + dy * dy + dz * dz) + 1e-12f);
    float Y[9];
    sh_l2(dx * inv, dy * inv, dz * inv, Y);
    #pragma unroll
    for (int k = 0; k < 16; ++k) {
        float h = 0.f;
        #pragma unroll
        for (int j = 0; j < 9; ++j) h = fmaf(sA1[j * 16 + k], Y[j], h);
        atomicAdd(&agg1[r * 16 + k], h);
    }
}

__global__ void __launch_bounds__(256) node_layer1(
    const float* __restrict__ agg1, const float* __restrict__ w1, const float* __restrict__ b1,
    float* __restrict__ x1, int n)
{
    int t = blockIdx.x * blockDim.x + threadIdx.x;
    if (t < n) x1[t] = fmaf(w1[0], agg1[t], b1[0]);
}

// ------------------------------------------------------------------
// Layer 2 as WMMA GEMM: per 16-edge tile, D(16x48) += U(16x144) * K2f(144x48)
// U[e, i*9+j] = x1[sender_e, i] * Y[e, j]; f32 16x16x4 WMMA, 108 per tile.
// ------------------------------------------------------------------
__global__ void __launch_bounds__(128) edge_layer2_wmma(
    const float* __restrict__ pos, const int* __restrict__ snd, const int* __restrict__ rcv,
    const float* __restrict__ x1, const float* __restrict__ Bswz,
    float* __restrict__ agg2, int E)
{
    __shared__ __align__(16) float Bl[36 * 3 * 32 * 2];  // 6912 floats, B in WMMA lane layout
    __shared__ __align__(16) float ed[4][16][29];        // per-wave: [0..15]=x1_s, [16..24]=Y
    __shared__ int rv[4][16];

    for (int i = threadIdx.x; i < 36 * 3 * 32 * 2; i += blockDim.x) Bl[i] = Bswz[i];
    __syncthreads();

    const int lane = threadIdx.x & 31;
    const int wid  = threadIdx.x >> 5;
    const int eL   = lane & 15;
    const bool hi  = (lane >= 16);
    const int nWaves = (gridDim.x * blockDim.x) >> 5;
    const int tiles  = E >> 4;
    const v2f* __restrict__ bp = (const v2f*)Bl;

    for (int tile = (blockIdx.x * blockDim.x + threadIdx.x) >> 5; tile < tiles; tile += nWaves) {
        int e0 = tile << 4;
        if (lane < 16) {
            int e = e0 + lane;
            int s = snd[e], r = rcv[e];
            rv[wid][lane] = r;
            float dx = pos[3 * r + 0] - pos[3 * s + 0];
            float dy = pos[3 * r + 1] - pos[3 * s + 1];
            float dz = pos[3 * r + 2] - pos[3 * s + 2];
            float inv = 1.0f / (sqrtf(dx * dx + dy * dy + dz * dz) + 1e-12f);
            float Y[9];
            sh_l2(dx * inv, dy * inv, dz * inv, Y);
            #pragma unroll
            for (int j = 0; j < 9; ++j) ed[wid][lane][16 + j] = Y[j];
            const float4* xs = (const float4*)(x1 + (size_t)s * 16);
            #pragma unroll
            for (int q = 0; q < 4; ++q) {
                float4 v = xs[q];
                ed[wid][lane][4 * q + 0] = v.x; ed[wid][lane][4 * q + 1] = v.y;
                ed[wid][lane][4 * q + 2] = v.z; ed[wid][lane][4 * q + 3] = v.w;
            }
            int nextE = e0 + (nWaves << 4);
            if (nextE + lane < E) {                     // global_prefetch_b8 path
                __builtin_prefetch(snd + nextE + lane, 0, 1);
                __builtin_prefetch(rcv + nextE + lane, 0, 1);
            }
        }
        __builtin_amdgcn_wave_barrier();  // LDS ops are in-order per wave; fence the compiler

        const float* __restrict__ my = &ed[wid][eL][0];
        v8f acc0 = {}, acc1 = {}, acc2 = {};
        #pragma unroll
        for (int step = 0; step < 36; ++step) {
            // A element indices (fold to literals per unrolled step, select on lane-half)
            int i0 = hi ? ((4 * step + 2) / 9) : ((4 * step + 0) / 9);
            int j0 = hi ? ((4 * step + 2) % 9) : ((4 * step + 0) % 9);
            int i1 = hi ? ((4 * step + 3) / 9) : ((4 * step + 1) / 9);
            int j1 = hi ? ((4 * step + 3) % 9) : ((4 * step + 1) % 9);
            v2f a;
            a.x = my[i0] * my[16 + j0];
            a.y = my[i1] * my[16 + j1];
            v2f b0 = bp[(step * 3 + 0) * 32 + lane];
            v2f b1 = bp[(step * 3 + 1) * 32 + lane];
            v2f b2 = bp[(step * 3 + 2) * 32 + lane];
            acc0 = __builtin_amdgcn_wmma_f32_16x16x4_f32(false, a, false, b0, (short)0, acc0, false, false);
            acc1 = __builtin_amdgcn_wmma_f32_16x16x4_f32(false, a, false, b1, (short)0, acc1, false, false);
            acc2 = __builtin_amdgcn_wmma_f32_16x16x4_f32(false, a, false, b2, (short)0, acc2, false, false);
        }
        __builtin_amdgcn_wave_barrier();

        // D layout: VGPR r -> lanes 0-15: (edge=r, k=lane); lanes 16-31: (edge=8+r, k=lane-16)
        int kk = lane & 15;
        #pragma unroll
        for (int rr = 0; rr < 8; ++rr) {
            int eIdx = hi ? (rr + 8) : rr;
            int node = rv[wid][eIdx];
            float* base = agg2 + (size_t)node * 36;
            atomicAdd(base + kk,        acc0[rr]);
            atomicAdd(base + 16 + kk,   acc1[rr]);
            if (kk < 4) atomicAdd(base + 32 + kk, acc2[rr]);
        }
    }
}

__global__ void __launch_bounds__(256) node_layer2_pool(
    const float* __restrict__ agg2, const float* __restrict__ w2, const float* __restrict__ b2,
    float* __restrict__ gsum, int N, int nodesPerGraph)
{
    int n = blockIdx.x * blockDim.x + threadIdx.x;
    if (n >= N) return;
    int g = n / nodesPerGraph;
    float w = w2[0], b = b2[0];
    #pragma unroll 4
    for (int k = 0; k < 36; ++k) {
        float v = fmaf(w, agg2[(size_t)n * 36 + k], b);
        atomicAdd(&gsum[g * 36 + k], v);
    }
}

// logits[g,c] = bg[c] + sum_k gsum[g,k] * (wg[k,c] + wg[36+k,c])  (parity channels equal)
__global__ void __launch_bounds__(256) readout(
    const float* __restrict__ gsum, const float* __restrict__ wg, const float* __restrict__ bg,
    float* __restrict__ out, int G)
{
    int t = blockIdx.x * blockDim.x + threadIdx.x;
    if (t >= G * 10) return;
    int g = t / 10, c = t % 10;
    float s = bg[c];
    #pragma unroll 4
    for (int k = 0; k < 36; ++k)
        s = fmaf(gsum[g * 36 + k], wg[k * 10 + c] + wg[(36 + k) * 10 + c], s);
    out[t] = s;
}

extern "C" void kernel_launch(void* const* d_in, const int* in_sizes, int n_in,
                              void* d_out, int out_size, void* d_ws, size_t ws_size,
                              hipStream_t stream) {
    const float* pos = (const float*)d_in[0];
    const float* w1  = (const float*)d_in[1];
    const float* b1  = (const float*)d_in[2];
    const float* w2  = (const float*)d_in[3];
    const float* b2  = (const float*)d_in[4];
    const float* wg  = (const float*)d_in[5];
    const float* bg  = (const float*)d_in[6];
    const int*   snd = (const int*)d_in[7];
    const int*   rcv = (const int*)d_in[8];

    const int N = in_sizes[0] / 3;
    const int E = in_sizes[7];
    const int G = out_size / 10;
    const int npg = N / G;

    char* ws = (char*)d_ws;
    float* A1   = (float*)(ws);            // 576 B
    float* Bswz = (float*)(ws + 1024);     // 27648 B
    float* agg1 = (float*)(ws + 32768);    // N*16*4
    float* x1   = agg1 + (size_t)N * 16;   // N*16*4
    float* agg2 = x1   + (size_t)N * 16;   // N*36*4
    float* gsum = agg2 + (size_t)N * 36;   // G*36*4

    hipMemsetAsync(agg1, 0, (size_t)N * 16 * sizeof(float), stream);
    hipMemsetAsync(agg2, 0, (size_t)N * 36 * sizeof(float), stream);
    hipMemsetAsync(gsum, 0, (size_t)G * 36 * sizeof(float), stream);

    setup_tables<<<28, 256, 0, stream>>>(A1, Bswz);
    edge_layer1<<<(E + 255) / 256, 256, 0, stream>>>(pos, snd, rcv, A1, agg1, E);
    node_layer1<<<(N * 16 + 255) / 256, 256, 0, stream>>>(agg1, w1, b1, x1, N * 16);
    edge_layer2_wmma<<<640, 128, 0, stream>>>(pos, snd, rcv, x1, Bswz, agg2, E);
    node_layer2_pool<<<(N + 255) / 256, 256, 0, stream>>>(agg2, w2, b2, gsum, N, npg);
    readout<<<(G * 10 + 255) / 256, 256, 0, stream>>>(gsum, wg, bg, (float*)d_out, G);
}